// SingleHeadNeuralMemoryMLP_26989574488237
// MI455X (gfx1250) — compile-verified
//
#include <hip/hip_runtime.h>
#include <hip/hip_bf16.h>

// ---------------------------------------------------------------------------
// SingleHeadNeuralMemoryMLP for MI455X (gfx1250, wave32)
//
// - All contractions go through V_WMMA_F32_16X16X4_F32 (fp32 end-to-end: the
//   cumulative-decay exponentials have large dynamic range, and fp32 WMMA
//   keeps the matrix pipes busy while staying bitwise-close to the reference).
// - K-tiles are staged global->LDS with GLOBAL_LOAD_ASYNC_TO_LDS_B128
//   (ASYNCcnt-tracked DMA, no VGPR round trip), double-buffered so the DMA of
//   tile i+1 overlaps the WMMA of tile i, with a single barrier per K-tile.
// - Elementwise work (silu, silu', bias, decay scaling, triangular mask,
//   -inner + cross, W decay) is fused into GEMM epilogues.
//
// Workspace requirement: ~193 MB (48.3M floats).
// ---------------------------------------------------------------------------

typedef float v2f __attribute__((ext_vector_type(2)));
typedef float v8f __attribute__((ext_vector_type(8)));

#define BM 64
#define BN 64
#define BK 16
#define APAD 20   // TA==0 A-tile row stride  (As[m][k])   : 16B-aligned, conflict-free reads
#define BPAD 68   // TB==0 B-tile row stride  (Bs[k][n])
#define BTP  68   // TA==1 A-tile row stride  (As[k][m], global-contiguous along m)
#define BKP  20   // TB==1 B-tile row stride  (Bs[n][k], global-contiguous along k)
#define SMEM_BYTES 20480  // max over layouts: 2*(64*20 + 64*20)*4

__device__ __forceinline__ float sigmoidf_(float x) { return 1.0f / (1.0f + expf(-x)); }
__device__ __forceinline__ float siluf_(float x)    { return x * sigmoidf_(x); }
__device__ __forceinline__ float silu_bwd_(float z) {
    float sg = sigmoidf_(z);
    float s  = z * sg;
    return s + sg * (1.0f - s);
}

__device__ __forceinline__ void async_copy16_(uint32_t lds_byte_off, const float* gaddr) {
    asm volatile("global_load_async_to_lds_b128 %0, %1, off"
                 :: "v"(lds_byte_off), "v"(gaddr) : "memory");
}

__device__ __forceinline__ void wait_async0_() {
    asm volatile("s_wait_asynccnt 0" ::: "memory");
}

struct EpiArgs {
    const float* p0;  long long p0_bs;    // aux matrix (bias / value / Z1 / inner / W_old)
    float*       out2; long long out2_bs; // second output (X2 = silu(Z1))
    const float* lr;                      // [B,L], batch stride L
    const float* logc;                    // [B,L] cumsum(log_wd), batch stride L
    int L;
};

// EPI codes:
// 0 store      : C = acc
// 1 bias       : C = acc + p0[n]
// 2 silu2      : C = acc ; out2 = silu(acc)
// 3 subv       : C = acc - p0[m,n]                      (grad_Z2 = Z2 - value)
// 4 silubwd    : C = silu_bwd(p0[m,n]) * acc            (grad_Z1)
// 5 scores     : C = (n<=m) ? acc*lr[n]*exp(logc[m]-logc[n]) : 0
// 6 cross      : C = acc*exp(logc[m]) - p0[m,n]         (Z2_ = cross - inner)
// 7 cross_silu : C = silu(acc*exp(logc[m]) - p0[m,n])   (X2_)
// 8 wnext      : C = -acc + p0[m,n]*exp(logc[L-1])      (W_next)
template <int TA, int TB, int EPI>
__global__ __launch_bounds__(256)
void gemm_wmma(const float* __restrict__ A, long long a_bs, int lda,
               const float* __restrict__ Bm, long long b_bs, int ldb,
               float* __restrict__ C, long long c_bs, int ldc,
               int M, int N, int K, int ktri, EpiArgs ea)
{
    extern __shared__ float smemf[];
    constexpr int ASZ = (TA == 0) ? BM * APAD : BK * BTP;   // floats per A buffer
    constexpr int BSZ = (TB == 0) ? BK * BPAD : BN * BKP;   // floats per B buffer
    // buffer f of A lives at float offset f*ASZ; buffer f of B at 2*ASZ + f*BSZ
    const uint32_t smem_base = (uint32_t)(uintptr_t)smemf;  // wave-relative LDS byte offset

    const int b  = blockIdx.z;
    const int m0 = blockIdx.y * BM;
    const int n0 = blockIdx.x * BN;

    A  += (long long)b * a_bs;
    Bm += (long long)b * b_bs;
    C  += (long long)b * c_bs;
    const float* lrp = ea.lr   ? ea.lr   + (long long)b * ea.L : nullptr;
    const float* lgp = ea.logc ? ea.logc + (long long)b * ea.L : nullptr;

    const int t    = threadIdx.x;
    const int lane = t & 31;
    const int wave = t >> 5;
    const int wm   = wave & 1;   // 2 wave rows -> 32 M each
    const int wn   = wave >> 1;  // 4 wave cols -> 16 N each
    const int lo   = lane & 15;
    const int hi   = lane >> 4;

    v8f acc[2];
    acc[0] = (v8f){0,0,0,0,0,0,0,0};
    acc[1] = (v8f){0,0,0,0,0,0,0,0};

    // fully-masked score blocks: min(l)=n0 > max(m)=m0+63 -> all zero, skip K loop
    if (EPI == 5 && n0 > m0 + (BM - 1)) {
        int n = n0 + wn * 16 + lo;
        #pragma unroll
        for (int s = 0; s < 2; ++s)
            #pragma unroll
            for (int r = 0; r < 8; ++r) {
                int m = m0 + wm * 32 + s * 16 + r + 8 * hi;
                C[(long long)m * ldc + n] = 0.0f;
            }
        return;
    }

    // triangular K clip for inner = P @ G : P[m,l]==0 for l>m
    int Keff = K;
    if (ktri) { int kl = m0 + BM; Keff = kl < K ? kl : K; }
    const int nIter = Keff / BK;

    // each wave DMAs 512B of A and 512B of B per tile (8 waves cover 4KB+4KB)
    auto issue_tile = [&](int buf, int k0) {
        {
            const float* g;
            uint32_t eoff;                        // float offset within A buffer
            if (TA == 0) {
                int row = wave * 8 + (lane >> 2);
                int ak  = (lane & 3) * 4;
                g    = A + (long long)(m0 + row) * lda + (k0 + ak);
                eoff = (uint32_t)(row * APAD + ak);
            } else {
                int k  = wave * 2 + (lane >> 4);
                int m4 = (lane & 15) * 4;
                g    = A + (long long)(k0 + k) * lda + (m0 + m4);
                eoff = (uint32_t)(k * BTP + m4);
            }
            async_copy16_(smem_base + (uint32_t)(buf * ASZ + eoff) * 4u, g);
        }
        {
            const float* g;
            uint32_t eoff;                        // float offset within B buffer
            if (TB == 0) {
                int k  = wave * 2 + (lane >> 4);
                int n4 = (lane & 15) * 4;
                g    = Bm + (long long)(k0 + k) * ldb + (n0 + n4);
                eoff = (uint32_t)(k * BPAD + n4);
            } else {
                int nn = wave * 8 + (lane >> 2);
                int k4 = (lane & 3) * 4;
                g    = Bm + (long long)(n0 + nn) * ldb + (k0 + k4);
                eoff = (uint32_t)(nn * BKP + k4);
            }
            async_copy16_(smem_base + (uint32_t)(2 * ASZ + buf * BSZ + eoff) * 4u, g);
        }
    };

    const int arow0 = wm * 32 + lo;
    const int bcol  = wn * 16 + lo;
    const int k2    = hi * 2;

    issue_tile(0, 0);
    for (int it = 0; it < nIter; ++it) {
        wait_async0_();        // this wave's share of tile `it` has landed in LDS
        __syncthreads();       // everyone's share has landed; prior tile fully consumed
        if (it + 1 < nIter) issue_tile((it + 1) & 1, (it + 1) * BK);  // overlaps compute

        const int cur = it & 1;
        const float* As = smemf + cur * ASZ;
        const float* Bs = smemf + 2 * ASZ + cur * BSZ;
        #pragma unroll
        for (int kk = 0; kk < BK; kk += 4) {
            const int kA = kk + k2;
            v2f a0, a1, bb;
            // A operand: lane = M row; vgpr0 holds K=2*hi, vgpr1 holds K=2*hi+1
            if (TA == 0) {
                a0.x = As[arow0 * APAD + kA];          a0.y = As[arow0 * APAD + kA + 1];
                a1.x = As[(arow0 + 16) * APAD + kA];   a1.y = As[(arow0 + 16) * APAD + kA + 1];
            } else {
                a0.x = As[kA * BTP + arow0];           a0.y = As[(kA + 1) * BTP + arow0];
                a1.x = As[kA * BTP + arow0 + 16];      a1.y = As[(kA + 1) * BTP + arow0 + 16];
            }
            // B operand: lane = N col; same K split across the two vgprs
            if (TB == 0) {
                bb.x = Bs[kA * BPAD + bcol];           bb.y = Bs[(kA + 1) * BPAD + bcol];
            } else {
                bb.x = Bs[bcol * BKP + kA];            bb.y = Bs[bcol * BKP + kA + 1];
            }
            acc[0] = __builtin_amdgcn_wmma_f32_16x16x4_f32(false, a0, false, bb,
                                                           (short)0, acc[0], false, false);
            acc[1] = __builtin_amdgcn_wmma_f32_16x16x4_f32(false, a1, false, bb,
                                                           (short)0, acc[1], false, false);
        }
    }

    // ---- fused epilogue ----
    const int n = n0 + wn * 16 + lo;
    const float* p0b = ea.p0   ? ea.p0   + (long long)b * ea.p0_bs   : nullptr;
    float*       o2b = ea.out2 ? ea.out2 + (long long)b * ea.out2_bs : nullptr;

    float bias = 0.0f, lrn = 0.0f, lgn = 0.0f, wdc_last = 0.0f;
    if (EPI == 1) bias = p0b[n];
    if (EPI == 5) { lrn = lrp[n]; lgn = lgp[n]; }
    if (EPI == 8) wdc_last = expf(lgp[ea.L - 1]);

    #pragma unroll
    for (int s = 0; s < 2; ++s) {
        #pragma unroll
        for (int r = 0; r < 8; ++r) {
            int m = m0 + wm * 32 + s * 16 + r + 8 * hi;     // C/D layout: M = r + 8*hi
            float v = acc[s][r];
            float o;
            if      (EPI == 0) o = v;
            else if (EPI == 1) o = v + bias;
            else if (EPI == 2) { o = v; o2b[(long long)m * ldc + n] = siluf_(v); }
            else if (EPI == 3) o = v - p0b[(long long)m * ldc + n];
            else if (EPI == 4) o = silu_bwd_(p0b[(long long)m * ldc + n]) * v;
            else if (EPI == 5) o = (n <= m) ? v * lrn * expf(lgp[m] - lgn) : 0.0f;
            else if (EPI == 6) o = v * expf(lgp[m]) - p0b[(long long)m * ldc + n];
            else if (EPI == 7) o = siluf_(v * expf(lgp[m]) - p0b[(long long)m * ldc + n]);
            else               o = -v + p0b[(long long)m * ldc + n] * wdc_last;
            C[(long long)m * ldc + n] = o;
        }
    }
}

// ---------------------------------------------------------------------------
// Per-row dot products: lr[b,l], log_wd[b,l]. One wave per row (wave32).
// ---------------------------------------------------------------------------
__global__ __launch_bounds__(256)
void row_stats_kernel(const float* __restrict__ x,
                      const float* __restrict__ w_lr, const float* __restrict__ b_lr,
                      const float* __restrict__ w_wd, const float* __restrict__ b_wd,
                      const float* __restrict__ lbl,  const float* __restrict__ lbw,
                      float* __restrict__ lr, float* __restrict__ logwd, int D)
{
    int wave = threadIdx.x >> 5, lane = threadIdx.x & 31;
    long long row = (long long)blockIdx.x * 8 + wave;
    const float* xr = x + row * D;
    float s1 = 0.0f, s2 = 0.0f;
    for (int i = lane; i < D; i += 32) {
        float xv = xr[i];
        s1 += xv * w_lr[i];
        s2 += xv * w_wd[i];
    }
    #pragma unroll
    for (int off = 16; off > 0; off >>= 1) {
        s1 += __shfl_down(s1, off, 32);
        s2 += __shfl_down(s2, off, 32);
    }
    if (lane == 0) {
        float a = s1 + b_lr[0];
        float c = s2 + b_wd[0];
        lr[row]    = expf(lbl[0]) * sigmoidf_(a);
        logwd[row] = logf(1.0f - expf(lbw[0]) * sigmoidf_(c));
    }
}

// ---------------------------------------------------------------------------
// Per-batch inclusive scan of log_wd -> logc (in place), then
// coefA[l] = lr[l] * exp(logc[L-1] - logc[l]).  One wave per batch.
// ---------------------------------------------------------------------------
__global__ void scan_kernel(const float* __restrict__ lr, float* __restrict__ logc,
                            float* __restrict__ coefA, int L)
{
    int lane = threadIdx.x;
    long long off = (long long)blockIdx.x * L;
    float carry = 0.0f;
    for (int i0 = 0; i0 < L; i0 += 32) {
        float v = logc[off + i0 + lane];
        #pragma unroll
        for (int d = 1; d < 32; d <<= 1) {
            float nv = __shfl_up(v, d, 32);
            if (lane >= d) v += nv;
        }
        v += carry;
        logc[off + i0 + lane] = v;
        carry = __shfl(v, 31, 32);        // broadcast running total
    }
    float last = carry;                   // == logc[L-1], all lanes
    for (int i = lane; i < L; i += 32)
        coefA[off + i] = lr[off + i] * expf(last - logc[off + i]);
}

// ---------------------------------------------------------------------------
// out[row, :] = in[row, :] * coef[row]  (vectorized float4; C multiple of 4)
// ---------------------------------------------------------------------------
__global__ __launch_bounds__(256)
void scale_rows_kernel(const float4* __restrict__ in, const float* __restrict__ coef,
                       float4* __restrict__ out, int Cq, long long total4)
{
    long long g = (long long)blockIdx.x * 256 + threadIdx.x;
    if (g >= total4) return;
    float c = coef[g / Cq];
    float4 v = in[g];
    v.x *= c; v.y *= c; v.z *= c; v.w *= c;
    out[g] = v;
}

// ---------------------------------------------------------------------------
static inline EpiArgs mkEpi() {
    EpiArgs e;
    e.p0 = nullptr;  e.p0_bs = 0;
    e.out2 = nullptr; e.out2_bs = 0;
    e.lr = nullptr;  e.logc = nullptr;
    e.L = 0;
    return e;
}

extern "C" void kernel_launch(void* const* d_in, const int* in_sizes, int n_in,
                              void* d_out, int out_size, void* d_ws, size_t ws_size,
                              hipStream_t stream) {
    (void)in_sizes; (void)n_in; (void)out_size; (void)ws_size;

    const int Bn = 4, L = 2048, D = 256, H = 512;
    typedef long long ll;
    const ll LD = (ll)L * D, LH = (ll)L * H, LL = (ll)L * L;
    const ll HD = (ll)H * D, DH = (ll)D * H;

    const float* x    = (const float*)d_in[0];
    const float* W1   = (const float*)d_in[1];
    const float* W2   = (const float*)d_in[2];
    const float* Wq   = (const float*)d_in[3];
    const float* bq   = (const float*)d_in[4];
    const float* Wk   = (const float*)d_in[5];
    const float* bk   = (const float*)d_in[6];
    const float* Wv   = (const float*)d_in[7];
    const float* bv   = (const float*)d_in[8];
    const float* w_lr = (const float*)d_in[9];
    const float* b_lr = (const float*)d_in[10];
    const float* w_wd = (const float*)d_in[11];
    const float* b_wd = (const float*)d_in[12];
    const float* lbl  = (const float*)d_in[13];
    const float* lbw  = (const float*)d_in[14];

    float* out   = (float*)d_out;
    float* outZ2 = out;                     // [B,L,D]
    float* outW1 = outZ2 + (ll)Bn * LD;     // [B,H,D]
    float* outW2 = outW1 + (ll)Bn * HD;     // [B,D,H]

    // workspace layout (48.3M floats = ~193 MB)
    float* ws     = (float*)d_ws;
    float* q      = ws;
    float* key    = q      + (ll)Bn * LD;
    float* val    = key    + (ll)Bn * LD;   // value; later reused for gZ2*coef
    float* Z1     = val    + (ll)Bn * LD;   // Z1;    later reused for gZ1*coef
    float* X2     = Z1     + (ll)Bn * LH;
    float* gZ2    = X2     + (ll)Bn * LH;
    float* gZ1    = gZ2    + (ll)Bn * LD;
    float* X2d    = gZ1    + (ll)Bn * LH;   // X2_
    float* inner1 = X2d    + (ll)Bn * LH;
    float* inner2 = inner1 + (ll)Bn * LH;
    float* P      = inner2 + (ll)Bn * LD;   // shared P1/P2 buffer [B,L,L]
    float* lr     = P      + (ll)Bn * LL;
    float* logc   = lr     + (ll)Bn * L;
    float* coefA  = logc   + (ll)Bn * L;

    dim3 blk(256);
    const int SB = SMEM_BYTES;

    // 0) per-row scalars
    row_stats_kernel<<<(Bn * L) / 8, blk, 0, stream>>>(x, w_lr, b_lr, w_wd, b_wd,
                                                       lbl, lbw, lr, logc, D);
    // 1) cumsum + coef
    scan_kernel<<<Bn, 32, 0, stream>>>(lr, logc, coefA, L);

    EpiArgs e;

    // 2) q/k/v = x @ W + b        (NN, bias epilogue; W shared across batch)
    {
        dim3 g(D / BN, L / BM, Bn);
        e = mkEpi(); e.p0 = bq; e.p0_bs = 0;
        gemm_wmma<0,0,1><<<g, blk, SB, stream>>>(x, LD, D, Wq, 0, D, q,   LD, D, L, D, D, 0, e);
        e.p0 = bk;
        gemm_wmma<0,0,1><<<g, blk, SB, stream>>>(x, LD, D, Wk, 0, D, key, LD, D, L, D, D, 0, e);
        e.p0 = bv;
        gemm_wmma<0,0,1><<<g, blk, SB, stream>>>(x, LD, D, Wv, 0, D, val, LD, D, L, D, D, 0, e);
    }
    // 3) Z1 = key @ W1^T ; X2 = silu(Z1)
    {
        dim3 g(H / BN, L / BM, Bn);
        e = mkEpi(); e.out2 = X2; e.out2_bs = LH;
        gemm_wmma<0,1,2><<<g, blk, SB, stream>>>(key, LD, D, W1, HD, D, Z1, LH, H, L, H, D, 0, e);
    }
    // 4) gZ2 = X2 @ W2^T - value
    {
        dim3 g(D / BN, L / BM, Bn);
        e = mkEpi(); e.p0 = val; e.p0_bs = LD;
        gemm_wmma<0,1,3><<<g, blk, SB, stream>>>(X2, LH, H, W2, DH, H, gZ2, LD, D, L, D, H, 0, e);
    }
    // 5) gZ1 = silu'(Z1) * (gZ2 @ W2)
    {
        dim3 g(H / BN, L / BM, Bn);
        e = mkEpi(); e.p0 = Z1; e.p0_bs = LH;
        gemm_wmma<0,0,4><<<g, blk, SB, stream>>>(gZ2, LD, D, W2, DH, H, gZ1, LH, H, L, H, D, 0, e);
    }
    // 5b) pre-scale for W-updates: Z1 <- gZ1*coef, val <- gZ2*coef (buffers now dead)
    scale_rows_kernel<<<(unsigned)((Bn * LH / 4 + 255) / 256), blk, 0, stream>>>(
        (const float4*)gZ1, coefA, (float4*)Z1, H / 4, Bn * LH / 4);
    scale_rows_kernel<<<(unsigned)((Bn * LD / 4 + 255) / 256), blk, 0, stream>>>(
        (const float4*)gZ2, coefA, (float4*)val, D / 4, Bn * LD / 4);

    // 6) P1[m,l] = (q[m].key[l]) * lr[l] * exp(logc[m]-logc[l]) * [l<=m]
    {
        dim3 g(L / BN, L / BM, Bn);
        e = mkEpi(); e.lr = lr; e.logc = logc; e.L = L;
        gemm_wmma<0,1,5><<<g, blk, SB, stream>>>(q, LD, D, key, LD, D, P, LL, L, L, L, D, 0, e);
    }
    // 7) inner1 = P1 @ gZ1   (triangular K clip)
    {
        dim3 g(H / BN, L / BM, Bn);
        e = mkEpi();
        gemm_wmma<0,0,0><<<g, blk, SB, stream>>>(P, LL, L, gZ1, LH, H, inner1, LH, H, L, H, L, 1, e);
    }
    // 8) X2_ = silu( (q @ W1^T)*wd_cross[m] - inner1 )
    {
        dim3 g(H / BN, L / BM, Bn);
        e = mkEpi(); e.p0 = inner1; e.p0_bs = LH; e.logc = logc; e.L = L;
        gemm_wmma<0,1,7><<<g, blk, SB, stream>>>(q, LD, D, W1, HD, D, X2d, LH, H, L, H, D, 0, e);
    }
    // 9) W1_next = -(gZ1*coef)^T @ key + W1 * wd_cross[L-1]
    {
        dim3 g(D / BN, H / BM, Bn);
        e = mkEpi(); e.p0 = W1; e.p0_bs = HD; e.logc = logc; e.L = L;
        gemm_wmma<1,0,8><<<g, blk, SB, stream>>>(Z1, LH, H, key, LD, D, outW1, HD, D, H, D, L, 0, e);
    }
    // 10) P2[m,l] = (X2_[m].X2[l]) * lr[l] * exp(logc[m]-logc[l]) * [l<=m]
    {
        dim3 g(L / BN, L / BM, Bn);
        e = mkEpi(); e.lr = lr; e.logc = logc; e.L = L;
        gemm_wmma<0,1,5><<<g, blk, SB, stream>>>(X2d, LH, H, X2, LH, H, P, LL, L, L, L, H, 0, e);
    }
    // 11) inner2 = P2 @ gZ2   (triangular K clip)
    {
        dim3 g(D / BN, L / BM, Bn);
        e = mkEpi();
        gemm_wmma<0,0,0><<<g, blk, SB, stream>>>(P, LL, L, gZ2, LD, D, inner2, LD, D, L, D, L, 1, e);
    }
    // 12) Z2_ = (X2_ @ W2^T)*wd_cross[m] - inner2      -> output
    {
        dim3 g(D / BN, L / BM, Bn);
        e = mkEpi(); e.p0 = inner2; e.p0_bs = LD; e.logc = logc; e.L = L;
        gemm_wmma<0,1,6><<<g, blk, SB, stream>>>(X2d, LH, H, W2, DH, H, outZ2, LD, D, L, D, H, 0, e);
    }
    // 13) W2_next = -(gZ2*coef)^T @ X2 + W2 * wd_cross[L-1]   -> output
    {
        dim3 g(H / BN, D / BM, Bn);
        e = mkEpi(); e.p0 = W2; e.p0_bs = DH; e.logc = logc; e.L = L;
        gemm_wmma<1,0,8><<<g, blk, SB, stream>>>(val, LD, D, X2, LH, H, outW2, DH, H, D, H, L, 0, e);
    }
}